// SUB_Loss_44796508897294
// MI455X (gfx1250) — compile-verified
//
#include <hip/hip_runtime.h>
#include <hip/hip_bf16.h>

#define BDIM 512
#define TEMP_INV (1.0f / 0.07f)

typedef float v2f __attribute__((ext_vector_type(2)));
typedef float v8f __attribute__((ext_vector_type(8)));

// -------------------------------------------------------------------------
// Kernel 1: sim = (A * A^T) / T  using V_WMMA_F32_16X16X4_F32.
// One wave (32 threads) computes one 16x16 output tile; grid = 32x32 tiles.
// A operand (16x4 fp32) lane layout:  lanes 0-15 hold M=0..15 (v0=K0,v1=K1),
// lanes 16-31 hold M=0..15 (v0=K2,v1=K3).  B = A^T, so B[k][n] = A[n][k],
// which gives the same per-lane load formula with the n-tile row base.
// -------------------------------------------------------------------------
__global__ __launch_bounds__(32) void sim_wmma_kernel(
    const float* __restrict__ A, float* __restrict__ sim) {
  const int lane = threadIdx.x;      // 0..31
  const int half = lane >> 4;        // 0 or 1
  const int l16  = lane & 15;
  const int mt = blockIdx.x * 16;
  const int nt = blockIdx.y * 16;

  const float* arow = A + (mt + l16) * BDIM;  // row M = mt + l16
  const float* brow = A + (nt + l16) * BDIM;  // row N = nt + l16 (B = A^T)

  v8f c = {};
  for (int k = 0; k < BDIM; k += 4) {
    // 8-byte aligned pair loads: {K+2*half, K+2*half+1}
    v2f a = *(const v2f*)(arow + k + 2 * half);
    v2f b = *(const v2f*)(brow + k + 2 * half);
    // (neg_a, A, neg_b, B, c_mod, C, reuse_a, reuse_b)
    c = __builtin_amdgcn_wmma_f32_16x16x4_f32(false, a, false, b,
                                              (short)0, c, false, false);
  }

  // C/D layout: VGPR v -> M = v (lanes 0-15) or v+8 (lanes 16-31), N = lane&15
#pragma unroll
  for (int v = 0; v < 8; ++v) {
    const int row = mt + v + 8 * half;
    const int col = nt + l16;
    sim[row * BDIM + col] = c[v] * TEMP_INV;
  }
}

// -------------------------------------------------------------------------
// Kernel 2: per-row masked pair loss.
// Block i handles row i.  Skips rows with mask_sents==0.  Builds compacted
// pos/neg index lists deterministically (thread 0 scans LDS-staged mask row),
// then all 256 threads sweep the pos x neg pair space with a fixed mapping.
// softplus(x) = max(x,0) + log1p(exp(-|x|))  (numerically stable).
// -------------------------------------------------------------------------
__global__ __launch_bounds__(256) void row_loss_kernel(
    const float* __restrict__ sim, const float* __restrict__ mask,
    const int* __restrict__ mask_sents, float* __restrict__ row_loss,
    unsigned* __restrict__ row_pairs) {
  const int i = blockIdx.x;
  const int t = threadIdx.x;

  __shared__ float    srow[BDIM];
  __shared__ float    mrow[BDIM];
  __shared__ short    poslist[BDIM];
  __shared__ short    neglist[BDIM];
  __shared__ unsigned cnt[2];
  __shared__ float    red[256];

  if (mask_sents[i] == 0) {
    if (t == 0) { row_loss[i] = 0.0f; row_pairs[i] = 0u; }
    return;
  }

  for (int j = t; j < BDIM; j += 256) {
    srow[j] = sim[i * BDIM + j];
    mrow[j] = mask[i * BDIM + j];
  }
  __syncthreads();

  if (t == 0) {  // deterministic compaction (512 LDS reads, negligible)
    unsigned np = 0, nn = 0;
    for (int j = 0; j < BDIM; ++j) {
      if (j == i) continue;
      if (mrow[j] > 0.5f) poslist[np++] = (short)j;
      else                neglist[nn++] = (short)j;
    }
    cnt[0] = np; cnt[1] = nn;
  }
  __syncthreads();

  const unsigned np = cnt[0];
  const unsigned nn = cnt[1];

  float acc = 0.0f;
  for (unsigned p = 0; p < np; ++p) {
    const float sp = srow[poslist[p]];            // LDS broadcast
    for (unsigned n = t; n < nn; n += 256) {
      const float x = srow[neglist[n]] - sp;      // sim[i,n] - sim[i,p]
      const float e = __expf(-fabsf(x));          // v_exp_f32
      acc += fmaxf(x, 0.0f) + __logf(1.0f + e);   // v_log_f32
    }
  }

  red[t] = acc;
  __syncthreads();
  for (int s = 128; s > 0; s >>= 1) {             // fixed-order tree
    if (t < s) red[t] += red[t + s];
    __syncthreads();
  }
  if (t == 0) {
    row_loss[i]  = red[0];
    row_pairs[i] = np * nn;                        // exact integer count
  }
}

// -------------------------------------------------------------------------
// Kernel 3: deterministic final reduction over 512 row partials + divide.
// -------------------------------------------------------------------------
__global__ __launch_bounds__(512) void finalize_kernel(
    const float* __restrict__ row_loss, const unsigned* __restrict__ row_pairs,
    float* __restrict__ out) {
  __shared__ float    rl[BDIM];
  __shared__ unsigned rp[BDIM];
  const int t = threadIdx.x;
  rl[t] = row_loss[t];
  rp[t] = row_pairs[t];
  __syncthreads();
  for (int s = 256; s > 0; s >>= 1) {
    if (t < s) { rl[t] += rl[t + s]; rp[t] += rp[t + s]; }
    __syncthreads();
  }
  if (t == 0) {
    const float loss_sum = rl[0];
    const float pair_num = (float)rp[0];
    out[0] = (pair_num > 0.0f) ? (loss_sum / fmaxf(pair_num, 1.0f)) : loss_sum;
  }
}

// -------------------------------------------------------------------------
extern "C" void kernel_launch(void* const* d_in, const int* in_sizes, int n_in,
                              void* d_out, int out_size, void* d_ws,
                              size_t ws_size, hipStream_t stream) {
  const float* features   = (const float*)d_in[0];  // (512,1,512) == (512,512)
  const float* mask       = (const float*)d_in[1];  // (512,512)
  const int*   mask_sents = (const int*)d_in[2];    // (512,)
  float* out = (float*)d_out;

  float*    sim       = (float*)d_ws;               // 512*512 f32 = 1 MB
  float*    row_loss  = sim + BDIM * BDIM;          // 512 f32
  unsigned* row_pairs = (unsigned*)(row_loss + BDIM); // 512 u32

  dim3 tiles(BDIM / 16, BDIM / 16);                 // 32 x 32 tiles
  sim_wmma_kernel<<<tiles, 32, 0, stream>>>(features, sim);
  row_loss_kernel<<<BDIM, 256, 0, stream>>>(sim, mask, mask_sents, row_loss,
                                            row_pairs);
  finalize_kernel<<<1, BDIM, 0, stream>>>(row_loss, row_pairs, out);
}